// MeshAttention_8100308320899
// MI455X (gfx1250) — compile-verified
//
#include <hip/hip_runtime.h>
#include <hip/hip_bf16.h>

#define KNN_K 16

typedef __attribute__((ext_vector_type(2))) float v2f;
typedef __attribute__((ext_vector_type(8))) float v8f;

// ---------------------------------------------------------------------------
// Kernel 1: exact KNN top-16 (by largest -dist^2, self included).
// One wave (32 lanes) per query point. Each lane keeps a sorted top-16 of its
// strided candidate subset; lanes merge with a log2(32)=5 step shfl_xor
// butterfly using the bitonic top-k merge (c[j] = max(a[j], b[15-j]) followed
// by a 4-stage bitonic re-sort). All register indices are static -> no spills.
// ---------------------------------------------------------------------------
__global__ __launch_bounds__(256) void knn_kernel(const float* __restrict__ xyz,
                                                  int* __restrict__ idx_out,
                                                  int N, int nWaves) {
  int wave = (blockIdx.x * blockDim.x + threadIdx.x) >> 5;
  int lane = threadIdx.x & 31;
  if (wave >= nWaves) return;
  int b = wave / N;
  int n = wave - b * N;
  const float* base = xyz + (size_t)b * N * 3;
  float qx = base[3 * n], qy = base[3 * n + 1], qz = base[3 * n + 2];
  float qn = qx * qx + qy * qy + qz * qz;

  float d[16];
  int id[16];
#pragma unroll
  for (int j = 0; j < 16; ++j) { d[j] = -3.0e38f; id[j] = 0; }

  for (int m = lane; m < N; m += 32) {
    float x = base[3 * m], y = base[3 * m + 1], z = base[3 * m + 2];
    float pd = 2.0f * (qx * x + qy * y + qz * z) - qn - (x * x + y * y + z * z);
    if (pd > d[15]) {              // rare after warm-up (~16/m probability)
      d[15] = pd; id[15] = m;
#pragma unroll
      for (int s = 15; s > 0; --s) {
        if (d[s] > d[s - 1]) {
          float td = d[s]; d[s] = d[s - 1]; d[s - 1] = td;
          int ti = id[s]; id[s] = id[s - 1]; id[s - 1] = ti;
        }
      }
    }
  }

  // butterfly merge across lanes; every lane converges to the wave's top-16
#pragma unroll
  for (int off = 16; off > 0; off >>= 1) {
    float od[16]; int oi[16];
#pragma unroll
    for (int j = 0; j < 16; ++j) {
      od[j] = __shfl_xor(d[j], off, 32);
      oi[j] = __shfl_xor(id[j], off, 32);
    }
    // top-16 of the union of two descending lists (bitonic first stage)
#pragma unroll
    for (int j = 0; j < 16; ++j) {
      float cd = od[15 - j]; int ci = oi[15 - j];
      if (cd > d[j]) { d[j] = cd; id[j] = ci; }
    }
    // bitonic merge: re-sort descending (stages 8,4,2,1)
#pragma unroll
    for (int s = 8; s > 0; s >>= 1) {
#pragma unroll
      for (int j = 0; j < 16; ++j) {
        if ((j & s) == 0) {
          if (d[j] < d[j + s]) {
            float td = d[j]; d[j] = d[j + s]; d[j + s] = td;
            int ti = id[j]; id[j] = id[j + s]; id[j + s] = ti;
          }
        }
      }
    }
  }

  if (lane == 0) {
#pragma unroll
    for (int j = 0; j < 16; ++j) idx_out[(size_t)wave * KNN_K + j] = id[j];
  }
}

// ---------------------------------------------------------------------------
// Kernel 2 (optional, if workspace is big enough): [B,C,N] -> [B,N,C] so the
// neighbor-feature gather reads coalesced 1KB rows instead of 16KB-strided
// scalars.
// ---------------------------------------------------------------------------
__global__ __launch_bounds__(256) void transpose_kernel(const float* __restrict__ in,
                                                        float* __restrict__ out,
                                                        int C, int N) {
  __shared__ float tile[32][33];
  int b = blockIdx.z;
  int n0 = blockIdx.x * 32;
  int c0 = blockIdx.y * 32;
  const float* src = in + (size_t)b * C * N;
  for (int r = threadIdx.y; r < 32; r += 8)
    tile[r][threadIdx.x] = src[(size_t)(c0 + r) * N + n0 + threadIdx.x];
  __syncthreads();
  float* dst = out + (size_t)b * N * C;
  for (int r = threadIdx.y; r < 32; r += 8)
    dst[(size_t)(n0 + r) * C + c0 + threadIdx.x] = tile[threadIdx.x][r];
}

// ---------------------------------------------------------------------------
// Kernel 3: fused projections + attention. One wave per 16-point tile.
//   query = cf_tile * Wq^T + bq           (WMMA f32 16x16x4, bias in C init)
//   qk    = query * Wk                    (WMMA f32 16x16x4)
//   score[n][k] = (qk[n] . (cf[idx]-cf[n]) + query[n].bk) / sqrt(128)
//   attn  = softmax_k(score)
//   out[b][c][n] = sum_k attn[n][k] * feats[b][c][idx[n][k]]
// ---------------------------------------------------------------------------
__global__ __launch_bounds__(32) void attn_kernel(
    const float* __restrict__ cf, const float* __restrict__ wq,
    const float* __restrict__ bq, const float* __restrict__ wk,
    const float* __restrict__ bk, const float* __restrict__ feats,
    const float* __restrict__ featsT, const int* __restrict__ knn,
    float* __restrict__ out, int N, int use_tr) {
  __shared__ float s_cf[16][128];
  __shared__ float s_q[16][128];
  __shared__ float s_qk[16][128];
  __shared__ float s_sc[16][16];
  __shared__ float s_at[16][16];
  __shared__ int   s_ix[16][16];
  __shared__ float s_qb[16];

  const int lane = threadIdx.x;
  const int tilesPerB = N / 16;
  const int b = blockIdx.x / tilesPerB;
  const int n0 = (blockIdx.x - b * tilesPerB) * 16;
  const int l16 = lane & 15;
  const int half = lane >> 4;   // A/B frags: lanes 0-15 hold K+{0,1}, 16-31 hold K+{2,3}
  const int koff = half * 2;

  // stage the 16x128 cf tile (contiguous 8KB) into LDS
  const float* cft = cf + ((size_t)b * N + n0) * 128;
  for (int t = lane; t < 16 * 128; t += 32)
    s_cf[t >> 7][t & 127] = cft[t];
  __syncthreads();

  // GEMM1: query(16x128) = cf_tile(16x128) * Wq^T(128x128) + bq
  for (int e0 = 0; e0 < 8; ++e0) {
    float bias = bq[e0 * 16 + l16];
    v8f acc;
#pragma unroll
    for (int v = 0; v < 8; ++v) acc[v] = bias;   // C[m][n] = bq[n]
    const float* wrow = wq + (size_t)(e0 * 16 + l16) * 128 + koff;
#pragma unroll
    for (int kk = 0; kk < 32; ++kk) {
      v2f a; a.x = s_cf[l16][4 * kk + koff]; a.y = s_cf[l16][4 * kk + koff + 1];
      v2f bf; bf.x = wrow[4 * kk]; bf.y = wrow[4 * kk + 1];
      acc = __builtin_amdgcn_wmma_f32_16x16x4_f32(false, a, false, bf,
                                                  (short)0, acc, false, false);
    }
#pragma unroll
    for (int v = 0; v < 8; ++v) s_q[v + 8 * half][e0 * 16 + l16] = acc[v];
  }
  __syncthreads();

  // GEMM2: qk(16x128) = query(16x128) * Wk(128x128)   (qk_d = sum_e q_e Wk[e][d])
  for (int d0 = 0; d0 < 8; ++d0) {
    v8f acc = {0, 0, 0, 0, 0, 0, 0, 0};
#pragma unroll
    for (int kk = 0; kk < 32; ++kk) {
      v2f a; a.x = s_q[l16][4 * kk + koff]; a.y = s_q[l16][4 * kk + koff + 1];
      const int e = 4 * kk + koff;
      v2f bf;
      bf.x = wk[(size_t)e * 128 + d0 * 16 + l16];
      bf.y = wk[(size_t)(e + 1) * 128 + d0 * 16 + l16];
      acc = __builtin_amdgcn_wmma_f32_16x16x4_f32(false, a, false, bf,
                                                  (short)0, acc, false, false);
    }
#pragma unroll
    for (int v = 0; v < 8; ++v) s_qk[v + 8 * half][d0 * 16 + l16] = acc[v];
  }
  if (lane < 16) {                       // q . bk term
    float s = 0.f;
    for (int e = 0; e < 128; ++e) s += s_q[lane][e] * bk[e];
    s_qb[lane] = s;
  }
  __syncthreads();

  // scores: 256 length-128 dot products (8 per lane), neighbor rows hit L2
  const int* knn_t = knn + ((size_t)b * N + n0) * KNN_K;
  const float* cfbatch = cf + (size_t)b * N * 128;
  const float inv_sqrt_dk = 0.08838834764831845f;  // 1/sqrt(128)
  for (int p = lane; p < 256; p += 32) {
    int nn = p >> 4, kk2 = p & 15;
    int j = knn_t[nn * KNN_K + kk2];
    s_ix[nn][kk2] = j;
    const float* cr = cfbatch + (size_t)j * 128;
    float s = 0.f;
#pragma unroll 4
    for (int dd = 0; dd < 128; ++dd)
      s += s_qk[nn][dd] * (cr[dd] - s_cf[nn][dd]);
    s_sc[nn][kk2] = (s + s_qb[nn]) * inv_sqrt_dk;
  }
  __syncthreads();

  // softmax over k=16 (lanes 0-15, one point each)
  if (lane < 16) {
    float mx = -3.0e38f;
#pragma unroll
    for (int j = 0; j < 16; ++j) mx = fmaxf(mx, s_sc[lane][j]);
    float sum = 0.f;
#pragma unroll
    for (int j = 0; j < 16; ++j) {
      float e = __expf(s_sc[lane][j] - mx);
      s_at[lane][j] = e;
      sum += e;
    }
    float inv = 1.0f / sum;
#pragma unroll
    for (int j = 0; j < 16; ++j) s_at[lane][j] *= inv;
  }
  __syncthreads();

  // out[b][c][n0+n] = sum_k attn * feat[idx]; lanes sweep c (8 chunks of 32)
  if (use_tr) {
    const float* ftb = featsT + (size_t)b * N * 256;
    for (int nn = 0; nn < 16; ++nn) {
      float acc[8] = {0, 0, 0, 0, 0, 0, 0, 0};
      for (int kk2 = 0; kk2 < 16; ++kk2) {
        float w = s_at[nn][kk2];
        const float* fr = ftb + (size_t)s_ix[nn][kk2] * 256;
#pragma unroll
        for (int q = 0; q < 8; ++q) acc[q] += w * fr[lane + 32 * q];  // coalesced
      }
#pragma unroll
      for (int q = 0; q < 8; ++q)
        out[((size_t)b * 256 + lane + 32 * q) * N + n0 + nn] = acc[q];
    }
  } else {
    const float* fb = feats + (size_t)b * 256 * N;
    for (int nn = 0; nn < 16; ++nn) {
      float acc[8] = {0, 0, 0, 0, 0, 0, 0, 0};
      for (int kk2 = 0; kk2 < 16; ++kk2) {
        float w = s_at[nn][kk2];
        int j = s_ix[nn][kk2];
#pragma unroll
        for (int q = 0; q < 8; ++q)
          acc[q] += w * fb[(size_t)(lane + 32 * q) * N + j];   // L2-resident
      }
#pragma unroll
      for (int q = 0; q < 8; ++q)
        out[((size_t)b * 256 + lane + 32 * q) * N + n0 + nn] = acc[q];
    }
  }
}

extern "C" void kernel_launch(void* const* d_in, const int* in_sizes, int n_in,
                              void* d_out, int out_size, void* d_ws, size_t ws_size,
                              hipStream_t stream) {
  (void)in_sizes; (void)n_in; (void)out_size;
  const float* xyz   = (const float*)d_in[0];  // [8,4096,3]
  const float* feats = (const float*)d_in[1];  // [8,256,4096]
  const float* cf    = (const float*)d_in[2];  // [8,4096,128]
  const float* wq    = (const float*)d_in[3];  // [128,128]
  const float* bq    = (const float*)d_in[4];  // [128]
  const float* wk    = (const float*)d_in[5];  // [128,128]
  const float* bk    = (const float*)d_in[6];  // [128]
  float* out = (float*)d_out;                  // [8,256,4096]

  const int B = 8, N = 4096, C = 256;

  size_t idx_bytes = (size_t)B * N * KNN_K * sizeof(int);           // 2 MB
  size_t idx_pad = (idx_bytes + 255) & ~(size_t)255;
  size_t tr_bytes = (size_t)B * N * C * sizeof(float);              // 32 MB
  int* d_idx = (int*)d_ws;
  float* d_ftr = (float*)((char*)d_ws + idx_pad);
  int use_tr = (ws_size >= idx_pad + tr_bytes) ? 1 : 0;  // constant per run -> deterministic

  int nWaves = B * N;                                    // 32768 waves
  knn_kernel<<<nWaves / 8, 256, 0, stream>>>(xyz, d_idx, N, nWaves);
  if (use_tr)
    transpose_kernel<<<dim3(N / 32, C / 32, B), dim3(32, 8), 0, stream>>>(feats, d_ftr, C, N);
  attn_kernel<<<B * (N / 16), 32, 0, stream>>>(cf, wq, bq, wk, bk, feats, d_ftr,
                                               d_idx, out, N, use_tr);
}